// LorentzLayer_80762565034115
// MI455X (gfx1250) — compile-verified
//
#include <hip/hip_runtime.h>

#define CLUSTER_ 100

typedef __attribute__((ext_vector_type(2))) float v2f;
typedef __attribute__((ext_vector_type(8))) float v8f;

// Standard Lorentz boost matrix for velocity b, metric (+,-,-,-):
//   B = I - g*|b|*(n.K) + (g-1)*(n.K)^2
// which closes to: B[0][0]=g, B[0][i+1]=B[i+1][0]=-g*b_i,
//                  B[i+1][k+1]=delta_ik + (g-1)*b_i*b_k/|b|^2
__device__ __forceinline__ void boost_mat(float b0, float b1, float b2, float m[4][4]) {
    float m2 = b0 * b0 + b1 * b1 + b2 * b2;
    float g  = rsqrtf(1.0f - m2);       // gamma = 1/sqrt(1-|b|^2)
    float f  = (g - 1.0f) / m2;
    m[0][0] = g;
    m[0][1] = m[1][0] = -g * b0;
    m[0][2] = m[2][0] = -g * b1;
    m[0][3] = m[3][0] = -g * b2;
    m[1][1] = 1.0f + f * b0 * b0;
    m[2][2] = 1.0f + f * b1 * b1;
    m[3][3] = 1.0f + f * b2 * b2;
    m[1][2] = m[2][1] = f * b0 * b1;
    m[1][3] = m[3][1] = f * b0 * b2;
    m[2][3] = m[3][2] = f * b1 * b2;
}

// Kernel 0: fuse M_c = Bi_mat @ (W_c * B_c) and emit it in the f32 4x16
// B-operand lane layout (lanes 0-15: K=0/1 in v0/v1; lanes 16-31: K=2/3),
// columns N>=4 zero-padded. Output: wsB[c*64 + vgpr*32 + lane].
__global__ void lorentz_prep(const float* __restrict__ Bo, const float* __restrict__ Bi,
                             const float* __restrict__ W, float* __restrict__ wsB) {
    int c = threadIdx.x;
    if (c >= CLUSTER_) return;
    float Mi[4][4], Bc[4][4], M[4][4];
    boost_mat(Bi[0], Bi[1], Bi[2], Mi);
    boost_mat(Bo[3 * c + 0], Bo[3 * c + 1], Bo[3 * c + 2], Bc);
    float w = W[c];
    for (int a = 0; a < 4; ++a)
        for (int d = 0; d < 4; ++d) {
            float s = 0.0f;
            for (int k = 0; k < 4; ++k) s += Mi[a][k] * Bc[k][d];
            M[a][d] = w * s;
        }
    // B[k][n] = M[n][k] (we need out = A * M^T); pad columns n>=4 with 0.
    for (int v = 0; v < 2; ++v)
        for (int lane = 0; lane < 32; ++lane) {
            int k = ((lane >= 16) ? 2 : 0) + v;
            int n = lane & 15;
            wsB[c * 64 + v * 32 + lane] = (n < 4) ? M[n][k] : 0.0f;
        }
}

// Kernel 1: streaming GEMM. Each wave owns 16 batch rows; per cluster:
// one global_load_b64 per lane (A operand, exactly in WMMA layout) and one
// v_wmma_f32_16x16x4_f32 accumulating into the 8-VGPR f32 accumulator.
__global__ __launch_bounds__(256) void lorentz_main(const float* __restrict__ T,
                                                    const float* __restrict__ wsB,
                                                    float* __restrict__ out) {
    __shared__ float ldsB[CLUSTER_ * 64];
    for (int i = threadIdx.x; i < CLUSTER_ * 64; i += 256) ldsB[i] = wsB[i];
    __syncthreads();

    const int lane = threadIdx.x & 31;
    const int wave = threadIdx.x >> 5;
    const int r0   = (blockIdx.x * 8 + wave) * 16;   // first batch row of this wave's tile
    const int n    = lane & 15;                      // row within tile (A) / column (B,D)
    const int koff = (lane >> 4) * 2;                // K pair held by this lane half

    const float* tp = T + (size_t)(r0 + n) * (CLUSTER_ * 4) + koff;

    v8f acc = {};
#pragma unroll 4
    for (int c = 0; c < CLUSTER_; ++c) {
        v2f a = *(const v2f*)(tp + c * 4);           // A frag: T[row, c, koff..koff+1]
        v2f b;
        b.x = ldsB[c * 64 + lane];                   // B frag v0
        b.y = ldsB[c * 64 + 32 + lane];              // B frag v1
        acc = __builtin_amdgcn_wmma_f32_16x16x4_f32(
            /*neg_a=*/false, a, /*neg_b=*/false, b,
            /*c_mod=*/(short)0, acc, /*reuse_a=*/false, /*reuse_b=*/false);
    }

    // D layout: VGPR v -> row v (lanes 0-15) or v+8 (lanes 16-31); lane&15 = col.
    if (n < 4) {
        int rb = r0 + ((lane >> 4) << 3);
#pragma unroll
        for (int v = 0; v < 8; ++v)
            out[(size_t)(rb + v) * 4 + n] = acc[v];
    }
}

extern "C" void kernel_launch(void* const* d_in, const int* in_sizes, int n_in,
                              void* d_out, int out_size, void* d_ws, size_t ws_size,
                              hipStream_t stream) {
    const float* T  = (const float*)d_in[0];   // (BATCH, 100, 4)
    const float* Bo = (const float*)d_in[1];   // (100, 3)
    const float* Bi = (const float*)d_in[2];   // (1, 3)
    const float* W  = (const float*)d_in[3];   // (100, 1)
    // d_in[4] = K_mats: generators are hardcoded in boost_mat.

    float* wsB = (float*)d_ws;                 // 100*64*4 = 25.6 KB of workspace
    float* out = (float*)d_out;

    const int batch = in_sizes[0] / (CLUSTER_ * 4);   // 262144

    lorentz_prep<<<1, 128, 0, stream>>>(Bo, Bi, W, wsB);

    const int rows_per_block = 128;                   // 8 waves * 16 rows
    lorentz_main<<<batch / rows_per_block, 256, 0, stream>>>(T, wsB, out);
}